// DigitCapsules_60455959658892
// MI455X (gfx1250) — compile-verified
//
#include <hip/hip_runtime.h>
#include <math.h>

typedef __attribute__((ext_vector_type(2))) float v2f;
typedef __attribute__((ext_vector_type(8))) float v8f;

#define R_TOT 4608
#define B_TOT 64
#define C_TOT 32
#define RCHUNK 16

// ---- WMMA helpers: D[16x16] = X[16x16] * W[16x16] via 4 chained 16x16x4 f32 WMMAs ----
// A frag (32-bit, 16x4): lane L holds A[M=L&15, K=2*(L>>4) + {0,1}] in {x,y}
// B frag (32-bit, 4x16): lane L holds B[K=2*(L>>4) + {0,1}, N=L&15] in {x,y}
// D frag (32-bit, 16x16): vgpr v, lane L -> D[M = v + 8*(L>>4), N = L&15]

__device__ __forceinline__ void load_a_frag(const float* __restrict__ x,
                                            int b0, int r, int lane, v2f a[4]) {
  int m = lane & 15, hl = lane >> 4;
  const float* px = x + ((size_t)(b0 + m) * R_TOT + r) * 16;
#pragma unroll
  for (int k = 0; k < 4; ++k) {
    int col = 4 * k + 2 * hl;
    v2f t; t.x = px[col]; t.y = px[col + 1];
    a[k] = t;
  }
}

// B fragment out of the LDS-staged W[r] slab (32 c-tiles of 256 floats)
__device__ __forceinline__ void load_b_frag_lds(const float* wst, int c, int lane, v2f b[4]) {
  int o = lane & 15, hl = lane >> 4;
  const float* pw = wst + (c << 8);
#pragma unroll
  for (int k = 0; k < 4; ++k) {
    int kk = 4 * k + 2 * hl;
    v2f t; t.x = pw[kk * 16 + o]; t.y = pw[kk * 16 + 16 + o];
    b[k] = t;
  }
}

__device__ __forceinline__ v8f wmma_chain(const v2f a[4], const v2f b[4]) {
  v8f acc = {0.f, 0.f, 0.f, 0.f, 0.f, 0.f, 0.f, 0.f};
#pragma unroll
  for (int k = 0; k < 4; ++k)
    acc = __builtin_amdgcn_wmma_f32_16x16x4_f32(false, a[k], false, b[k],
                                                (short)0, acc, false, false);
  return acc;
}

// ---- One routing pass: 8 waves/block; waves (bt, cw) = b-tile bt, capsule-half cw ----
// Dynamic LDS layout (floats):
//   sacc   : [4][32][16][16] @0      (32768)  s accumulator (exclusive owners)
//   wstage : [32][16][16]    @32768  (8192)   async-staged W[r] slab (shared by all waves)
//   vsl    : [4][32][16][16] @40960  (32768)  vsum staged per b-tile (softmax only)
//   tile   : [8][16][16]     @73728  (2048)   wave-private u_hat scratch for dots
//   lgt    : [4][32][16]     @75776  (2048)   logits per b-tile
//   wgt    : [4][32][16]     @77824  (2048)   softmax weights per b-tile
// totals: softmax 79872 floats (312 KB), uniform 40960 floats (160 KB)
__global__ void __launch_bounds__(256)
caps_pass(const float* __restrict__ x, const float* __restrict__ w,
          const float* __restrict__ vsum, float* __restrict__ s, int useSoftmax) {
  extern __shared__ float smem[];
  float* sacc   = smem;
  float* wstage = smem + 32768;
  float* vsl    = smem + 40960;
  float* tile   = smem + 73728;
  float* lgt    = smem + 75776;
  float* wgt    = smem + 77824;

  const int lane = threadIdx.x & 31;
  const int wv   = threadIdx.x >> 5;   // 0..7
  const int bt   = wv >> 1;            // b-tile 0..3
  const int cw   = wv & 1;             // capsule half 0..1
  const int cbase = cw * 16;
  const int ln   = lane & 15;
  const int hl   = lane >> 4;
  const int b0   = bt * 16;
  const int r0   = blockIdx.x * RCHUNK;

  for (int i = threadIdx.x; i < 32768; i += 256) sacc[i] = 0.f;
  if (useSoftmax) {
    // stage vsum[b][c][o] -> vsl[bt][c][m][o]
    for (int i = threadIdx.x; i < 32768; i += 256) {
      int o = i & 15, c = (i >> 4) & 31, b = i >> 9;
      vsl[(((b >> 4) * 32 + c) * 16 + (b & 15)) * 16 + o] = vsum[i];
    }
  }
  __syncthreads();

  const unsigned wstage_byte = 32768u * 4u;   // byte offset of wstage in dynamic LDS

  for (int rr = 0; rr < RCHUNK; ++rr) {
    const int r = r0 + rr;

    // ---- stage W[r] (32 KB, all 32 capsule tiles) into LDS via async copy engine
    {
      const float* gsrc = w + ((size_t)r << 15);       // W[r][0][0][0]
#pragma unroll
      for (int i = 0; i < 8; ++i) {
        unsigned fofs = (unsigned)(wv * 1024 + i * 128 + lane * 4);
        const float* ga = gsrc + fofs;
        unsigned la = wstage_byte + fofs * 4u;
        asm volatile("global_load_async_to_lds_b128 %0, %1, off"
                     :: "v"(la), "v"(ga) : "memory");
      }
      asm volatile("s_wait_asynccnt 0x0" ::: "memory");
    }
    __syncthreads();   // slab visible to all waves

    v2f a[4];
    load_a_frag(x, b0, r, lane, a);

    if (useSoftmax) {
      // ---- phase 1: logits[b,c] = <u_hat[b,r,c,:], vsum[b,c,:]> for this wave's 16 c
      for (int cc = 0; cc < 16; ++cc) {
        const int c = cbase + cc;
        v2f bf[4];
        load_b_frag_lds(wstage, c, lane, bf);
        v8f d = wmma_chain(a, bf);
#pragma unroll
        for (int v = 0; v < 8; ++v)
          tile[wv * 256 + (v + 8 * hl) * 16 + ln] = d[v];
        asm volatile("s_wait_dscnt 0" ::: "memory");
        // lanes (hl,ln): row m=ln, o-half hl; vectorized b128 LDS reads
        const float4* tr = (const float4*)(tile + wv * 256 + ln * 16 + hl * 8);
        const float4* vr = (const float4*)(vsl + ((bt * 32 + c) * 16 + ln) * 16 + hl * 8);
        float4 t0 = tr[0], t1 = tr[1], v0 = vr[0], v1 = vr[1];
        float p = t0.x * v0.x;
        p = fmaf(t0.y, v0.y, p); p = fmaf(t0.z, v0.z, p); p = fmaf(t0.w, v0.w, p);
        p = fmaf(t1.x, v1.x, p); p = fmaf(t1.y, v1.y, p);
        p = fmaf(t1.z, v1.z, p); p = fmaf(t1.w, v1.w, p);
        p += __shfl_xor(p, 16);
        lgt[(bt * 32 + c) * 16 + ln] = p;   // both halves write same value (no exec branch)
      }
      __syncthreads();

      // ---- phase 2: softmax over all 32 c; this wave handles rows m = cw*8 + (lane&7)
      {
        const int mrow = cw * 8 + (lane & 7);
        const int cg = lane >> 3;            // 0..3, 8 capsules each
        float l[8], mx = -3.0e38f;
#pragma unroll
        for (int j = 0; j < 8; ++j) {
          l[j] = lgt[(bt * 32 + cg * 8 + j) * 16 + mrow];
          mx = fmaxf(mx, l[j]);
        }
        mx = fmaxf(mx, __shfl_xor(mx, 8));
        mx = fmaxf(mx, __shfl_xor(mx, 16));
        float se = 0.f;
#pragma unroll
        for (int j = 0; j < 8; ++j) { l[j] = __expf(l[j] - mx); se += l[j]; }
        se += __shfl_xor(se, 8);
        se += __shfl_xor(se, 16);
        float inv = 1.f / se;
#pragma unroll
        for (int j = 0; j < 8; ++j)
          wgt[(bt * 32 + cg * 8 + j) * 16 + mrow] = l[j] * inv;
      }
      __syncthreads();
    }

    // ---- phase 3: sacc[b,c,:] += w_c * u_hat[b,r,c,:] (recompute via WMMA from LDS)
    for (int cc = 0; cc < 16; ++cc) {
      const int c = cbase + cc;
      v2f bf[4];
      load_b_frag_lds(wstage, c, lane, bf);
      v8f d = wmma_chain(a, bf);
      float wl[8];
      if (useSoftmax) {
        const float4* wq = (const float4*)(wgt + (bt * 32 + c) * 16 + 8 * hl);
        float4 w0 = wq[0], w1 = wq[1];
        wl[0] = w0.x; wl[1] = w0.y; wl[2] = w0.z; wl[3] = w0.w;
        wl[4] = w1.x; wl[5] = w1.y; wl[6] = w1.z; wl[7] = w1.w;
      } else {
#pragma unroll
        for (int v = 0; v < 8; ++v) wl[v] = 1.f / 32.f;
      }
#pragma unroll
      for (int v = 0; v < 8; ++v) {
        int idx = ((bt * 32 + c) * 16 + (v + 8 * hl)) * 16 + ln; // exclusive owner
        sacc[idx] += wl[v] * d[v];
      }
    }
    __syncthreads();   // protect wstage before next r's staging
  }

  // flush per-WG partial s to global with float atomics
  for (int i = threadIdx.x; i < 32768; i += 256) {
    int o = i & 15, mm = (i >> 4) & 15, c = (i >> 8) & 31, b2 = i >> 13;
    int b = b2 * 16 + mm;
    atomicAdd(&s[(b * 32 + c) * 16 + o], sacc[i]);
  }
}

// ---- squash + accumulate vsum; optionally emit final output ----
__global__ void __launch_bounds__(256)
caps_squash(const float* __restrict__ s, float scale, float* __restrict__ vsum,
            float* __restrict__ out, int writeOut) {
  int idx = blockIdx.x * 256 + threadIdx.x; // (b,c) pair, 2048 total
  if (idx >= B_TOT * C_TOT) return;
  float v[16], n2 = 0.f;
#pragma unroll
  for (int o = 0; o < 16; ++o) { v[o] = s[idx * 16 + o] * scale; n2 = fmaf(v[o], v[o], n2); }
  float n = sqrtf(n2);
  float f = (n2 / (1.f + n2)) / (n + 1e-8f);
#pragma unroll
  for (int o = 0; o < 16; ++o) {
    float val = f * v[o];
    vsum[idx * 16 + o] += val;
    if (writeOut) out[idx * 16 + o] = val;
  }
}

extern "C" void kernel_launch(void* const* d_in, const int* in_sizes, int n_in,
                              void* d_out, int out_size, void* d_ws, size_t ws_size,
                              hipStream_t stream) {
  const float* x = (const float*)d_in[0];          // [64,4608,16]
  const float* w = (const float*)d_in[1];          // [4608,32,16,16]
  float* out  = (float*)d_out;                     // [64,32,16]
  float* s    = (float*)d_ws;                      // 32768 floats
  float* vsum = s + 32768;                         // 32768 floats

  hipMemsetAsync(d_ws, 0, (size_t)2 * 32768 * sizeof(float), stream);

  dim3 grid(R_TOT / RCHUNK), blk(256);
  size_t smemU = (size_t)40960 * sizeof(float);    // uniform pass: sacc + wstage (160 KB)
  size_t smemS = (size_t)79872 * sizeof(float);    // softmax pass (312 KB)

  // iteration 1: uniform coupling (softmax of zeros == 1/32)
  caps_pass<<<grid, blk, smemU, stream>>>(x, w, nullptr, s, 0);
  caps_squash<<<8, 256, 0, stream>>>(s, 1.f / 32.f, vsum, out, 0);

  // iteration 2: weights = softmax_c(<u_hat, v1>)
  hipMemsetAsync(s, 0, (size_t)32768 * sizeof(float), stream);
  caps_pass<<<grid, blk, smemS, stream>>>(x, w, vsum, s, 1);
  caps_squash<<<8, 256, 0, stream>>>(s, 1.f, vsum, out, 0);

  // iteration 3: weights = softmax_c(<u_hat, v1+v2>), emit v3
  hipMemsetAsync(s, 0, (size_t)32768 * sizeof(float), stream);
  caps_pass<<<grid, blk, smemS, stream>>>(x, w, vsum, s, 1);
  caps_squash<<<8, 256, 0, stream>>>(s, 1.f, vsum, out, 1);
}